// MultiBoxLoss_65472481460366
// MI455X (gfx1250) — compile-verified
//
#include <hip/hip_runtime.h>

#define THRESH 0.5f
#define VAR0 0.1f
#define VAR1 0.2f
#define MAXO 64       // O = 50
#define MAXP 8960     // P = 8732
#define NCHUNK 21     // ceil(C/4), C = 81

typedef __attribute__((ext_vector_type(2))) float v2f;
typedef __attribute__((ext_vector_type(8))) float v8f;

// ---------------------------------------------------------------------------
// Kernel 1: per-batch matching + localization smooth-L1 loss
// ---------------------------------------------------------------------------
__global__ __launch_bounds__(256) void k_match(
    const float* __restrict__ targets,  // [B,O,5]
    const float* __restrict__ priors,   // [P,4] center-size
    const float* __restrict__ loc,      // [B,P,4]
    float* __restrict__ btOv, int* __restrict__ btIdx,
    int* __restrict__ confT, int* __restrict__ numPos,
    float* __restrict__ lossL,
    int P, int O)
{
    __shared__ float tb[MAXO][4];
    __shared__ float tl[MAXO];
    __shared__ unsigned long long bestT[MAXO];
    __shared__ float rf[256];
    __shared__ int   ri[256];

    const int b = blockIdx.x, tid = threadIdx.x;

    for (int i = tid; i < O * 5; i += 256) {
        int o = i / 5, j = i % 5;
        float vv = targets[(size_t)(b * O + o) * 5 + j];
        if (j < 4) tb[o][j] = vv; else tl[o] = vv;
    }
    for (int o = tid; o < O; o += 256) bestT[o] = 0ull;
    __syncthreads();

    // Pass 1: best truth per prior; best prior per truth (packed u64 max)
    for (int p = tid; p < P; p += 256) {
        float pcx = priors[p*4+0], pcy = priors[p*4+1];
        float pw  = priors[p*4+2], ph  = priors[p*4+3];
        float px1 = pcx - 0.5f*pw, py1 = pcy - 0.5f*ph;
        float px2 = pcx + 0.5f*pw, py2 = pcy + 0.5f*ph;
        float pA  = (px2-px1)*(py2-py1);
        float bov = -1.0f; int bo = 0;
        for (int o = 0; o < O; ++o) {
            float x1 = tb[o][0], y1 = tb[o][1], x2 = tb[o][2], y2 = tb[o][3];
            float iw = fmaxf(fminf(x2, px2) - fmaxf(x1, px1), 0.0f);
            float ih = fmaxf(fminf(y2, py2) - fmaxf(y1, py1), 0.0f);
            float inter = iw * ih;
            float iou = inter / ((x2-x1)*(y2-y1) + pA - inter);
            if (iou > bov) { bov = iou; bo = o; }   // first max -> lowest truth idx
            // tie-break toward lowest prior index via complemented index
            unsigned long long key =
                ((unsigned long long)__float_as_uint(iou) << 32) |
                (unsigned long long)(0xFFFFFFFFu - (unsigned)p);
            if (key > bestT[o]) atomicMax(&bestT[o], key);
        }
        size_t idx = (size_t)b * P + p;
        btOv[idx] = bov; btIdx[idx] = bo;
    }
    __syncthreads();

    // Pass 2: force each truth's best prior
    for (int o = tid; o < O; o += 256) {
        int po = (int)(0xFFFFFFFFu - (unsigned)(bestT[o] & 0xFFFFFFFFull));
        size_t idx = (size_t)b * P + po;
        btOv[idx] = 2.0f; btIdx[idx] = o;
    }
    __syncthreads();

    // Pass 3: conf targets + smooth-L1 over positives
    float ls = 0.0f; int pc = 0;
    for (int p = tid; p < P; p += 256) {
        size_t idx = (size_t)b * P + p;
        float ov = btOv[idx]; int o = btIdx[idx];
        int cf = (ov < THRESH) ? 0 : ((int)tl[o] + 1);
        confT[idx] = cf;
        if (cf > 0) {
            ++pc;
            float x1 = tb[o][0], y1 = tb[o][1], x2 = tb[o][2], y2 = tb[o][3];
            float pcx = priors[p*4+0], pcy = priors[p*4+1];
            float pw  = priors[p*4+2], ph  = priors[p*4+3];
            float g0 = ((x1+x2)*0.5f - pcx) / (VAR0 * pw);
            float g1 = ((y1+y2)*0.5f - pcy) / (VAR0 * ph);
            float g2 = __logf((x2-x1) / pw) / VAR1;
            float g3 = __logf((y2-y1) / ph) / VAR1;
            const float* ld = loc + idx * 4;
            float g[4] = {g0, g1, g2, g3};
            #pragma unroll
            for (int j = 0; j < 4; ++j) {
                float d = ld[j] - g[j];
                float ad = fabsf(d);
                ls += (ad < 1.0f) ? 0.5f*d*d : ad - 0.5f;
            }
        }
    }
    rf[tid] = ls; ri[tid] = pc; __syncthreads();
    for (int s = 128; s > 0; s >>= 1) {
        if (tid < s) { rf[tid] += rf[tid+s]; ri[tid] += ri[tid+s]; }
        __syncthreads();
    }
    if (tid == 0) { lossL[b] = rf[0]; numPos[b] = ri[0]; }
}

// ---------------------------------------------------------------------------
// Kernel 2: CE = logsumexp(conf_row) - conf_row[target], WMMA row-sum engine.
// One wave handles 16 rows; sum of exps via chained V_WMMA_F32_16X16X4_F32
// with an all-ones B matrix (D[i][j] = sum_k A[i][k], accumulated over 21
// K-chunks covering 84 >= C columns, pads contribute exp(-inf)=0).
// ---------------------------------------------------------------------------
__global__ __launch_bounds__(256) void k_ce(
    const float* __restrict__ conf, const int* __restrict__ confT,
    float* __restrict__ ce, int R, int C)
{
    const int lane = threadIdx.x & 31;
    const int wave = threadIdx.x >> 5;
    const int rowTile = (blockIdx.x * 8 + wave) * 16;
    const int row = rowTile + (lane & 15);
    const bool valid = row < R;                // uniform per wave (R % 16 == 0)
    const int colBase = (lane >> 4) * 2;       // A 16x4 f32 layout: lanes0-15 K=0,1; lanes16-31 K=2,3
    const float NINF = -__builtin_inff();

    float ax[NCHUNK], ay[NCHUNK];
    const float* rp = conf + (size_t)(valid ? row : 0) * C;
    #pragma unroll
    for (int c = 0; c < NCHUNK; ++c) {
        int c0 = 4*c + colBase;
        ax[c] = (valid && c0     < C) ? rp[c0]     : NINF;
        ay[c] = (valid && c0 + 1 < C) ? rp[c0 + 1] : NINF;
    }
    float m = NINF;
    #pragma unroll
    for (int c = 0; c < NCHUNK; ++c) { m = fmaxf(m, ax[c]); m = fmaxf(m, ay[c]); }
    m = fmaxf(m, __shfl_xor(m, 16, 32));       // combine the two half-wave K groups

    v2f ones; ones.x = 1.0f; ones.y = 1.0f;
    v8f acc = {0.f,0.f,0.f,0.f,0.f,0.f,0.f,0.f};
    #pragma unroll
    for (int c = 0; c < NCHUNK; ++c) {
        v2f av;
        av.x = __expf(ax[c] - m);
        av.y = __expf(ay[c] - m);
        acc = __builtin_amdgcn_wmma_f32_16x16x4_f32(
                  false, av, false, ones, (short)0, acc, false, false);
    }

    // Extract row sums from 16x16 f32 D layout: VGPR r holds M=r (lanes0-15) / M=r+8 (lanes16-31)
    int e3 = lane & 7;
    float s = acc[0];
    #pragma unroll
    for (int j = 1; j < 8; ++j) if (e3 == j) s = acc[j];
    float other = __shfl_xor(s, 16, 32);
    bool wantHi = (lane & 15) >= 8;
    bool isHi   = lane >= 16;
    float rs = (wantHi == isHi) ? s : other;

    if (valid && lane < 16) {
        int t = confT[row];
        ce[row] = (m + __logf(rs)) - rp[t];
    }
}

// ---------------------------------------------------------------------------
// Kernel 3: hard-negative mining = top-k sum of pos-masked CE (k = min(3*np,P-1))
// via 31-step binary search on non-negative float bit patterns in LDS.
// ---------------------------------------------------------------------------
__global__ __launch_bounds__(256) void k_mine(
    const float* __restrict__ ce, const int* __restrict__ confT,
    const int* __restrict__ numPos, float* __restrict__ lossC, int P)
{
    __shared__ float v[MAXP];
    __shared__ float rf[256];
    __shared__ int   ri[256];

    const int b = blockIdx.x, tid = threadIdx.x;
    float posSum = 0.0f;
    for (int p = tid; p < P; p += 256) {
        size_t idx = (size_t)b * P + p;
        float cv = ce[idx];
        bool ip = confT[idx] > 0;
        posSum += ip ? cv : 0.0f;
        v[p] = ip ? 0.0f : fmaxf(cv, 0.0f);    // clamp: CE >= 0 analytically
    }
    rf[tid] = posSum; __syncthreads();
    for (int s = 128; s > 0; s >>= 1) {
        if (tid < s) rf[tid] += rf[tid+s];
        __syncthreads();
    }
    float posTotal = rf[0];
    int np = numPos[b];
    int k  = min(3 * np, P - 1);
    __syncthreads();
    if (k <= 0) { if (tid == 0) lossC[b] = posTotal; return; }  // uniform branch

    // ans = bit pattern of k-th largest value: max u with #{bits(v) >= u} >= k
    unsigned ans = 0u;
    for (int bit = 30; bit >= 0; --bit) {
        unsigned cand = ans | (1u << bit);
        int cnt = 0;
        for (int p = tid; p < P; p += 256)
            cnt += (__float_as_uint(v[p]) >= cand) ? 1 : 0;
        ri[tid] = cnt; __syncthreads();
        for (int s = 128; s > 0; s >>= 1) {
            if (tid < s) ri[tid] += ri[tid+s];
            __syncthreads();
        }
        if (ri[0] >= k) ans = cand;            // same on every thread
        __syncthreads();
    }
    float T = __uint_as_float(ans);
    float sgt = 0.0f; int cgt = 0;
    for (int p = tid; p < P; p += 256) {
        if (__float_as_uint(v[p]) > ans) { sgt += v[p]; ++cgt; }
    }
    rf[tid] = sgt; ri[tid] = cgt; __syncthreads();
    for (int s = 128; s > 0; s >>= 1) {
        if (tid < s) { rf[tid] += rf[tid+s]; ri[tid] += ri[tid+s]; }
        __syncthreads();
    }
    if (tid == 0) lossC[b] = posTotal + rf[0] + (float)(k - ri[0]) * T;
}

// ---------------------------------------------------------------------------
// Kernel 4: deterministic final reduction over B=64 per-batch partials
// ---------------------------------------------------------------------------
__global__ void k_final(const float* __restrict__ lossL,
                        const float* __restrict__ lossC,
                        const int* __restrict__ numPos,
                        float* __restrict__ out, int B)
{
    if (threadIdx.x == 0 && blockIdx.x == 0) {
        float sl = 0.0f, sc = 0.0f; int n = 0;
        for (int b = 0; b < B; ++b) { sl += lossL[b]; sc += lossC[b]; n += numPos[b]; }
        float N = (float)n;
        out[0] = sl / N;
        out[1] = sc / N;
    }
}

// ---------------------------------------------------------------------------
extern "C" void kernel_launch(void* const* d_in, const int* in_sizes, int n_in,
                              void* d_out, int out_size, void* d_ws, size_t ws_size,
                              hipStream_t stream) {
    const float* loc_data  = (const float*)d_in[0];   // [B,P,4]
    const float* conf_data = (const float*)d_in[1];   // [B,P,C]
    const float* targets   = (const float*)d_in[2];   // [B,O,5]
    const float* priors    = (const float*)d_in[3];   // [P,4]

    const int P = in_sizes[3] / 4;
    const int B = in_sizes[0] / (4 * P);
    const int C = in_sizes[1] / (B * P);
    const int O = in_sizes[2] / (5 * B);
    const long long R = (long long)B * P;

    // workspace layout
    char* w = (char*)d_ws;
    float* btOv   = (float*)w;                 w += (size_t)R * sizeof(float);
    int*   btIdx  = (int*)w;                   w += (size_t)R * sizeof(int);
    int*   confT  = (int*)w;                   w += (size_t)R * sizeof(int);
    float* ceBuf  = (float*)w;                 w += (size_t)R * sizeof(float);
    int*   nPos   = (int*)w;                   w += (size_t)B * sizeof(int);
    float* lossLb = (float*)w;                 w += (size_t)B * sizeof(float);
    float* lossCb = (float*)w;                 w += (size_t)B * sizeof(float);

    k_match<<<B, 256, 0, stream>>>(targets, priors, loc_data,
                                   btOv, btIdx, confT, nPos, lossLb, P, O);

    int tiles = (int)((R + 127) / 128);        // 8 waves/block, 16 rows/wave
    k_ce<<<tiles, 256, 0, stream>>>(conf_data, confT, ceBuf, (int)R, C);

    k_mine<<<B, 256, 0, stream>>>(ceBuf, confT, nPos, lossCb, P);

    k_final<<<1, 32, 0, stream>>>(lossLb, lossCb, nPos, (float*)d_out, B);
}